// GCN_14242111553926
// MI455X (gfx1250) — compile-verified
//
#include <hip/hip_runtime.h>
#include <stdint.h>

typedef __attribute__((ext_vector_type(2))) float v2f;
typedef __attribute__((ext_vector_type(8))) float v8f;
typedef __attribute__((ext_vector_type(4))) unsigned int v4u;
typedef __attribute__((ext_vector_type(8))) int v8i;
typedef __attribute__((ext_vector_type(4))) int v4i;

#define N_NODES 50000
#define N_EDGES 625000
#define CH      128
#define LDS_STRIDE 130     // 128 + 2 pad (TDM pad_interval=128 dwords, pad_amount=2 dwords)
#define ROWS_PER_BLK 80    // 625 blocks * 80 rows = 50000; 5 x 16-row WMMA tiles
#define M_TILES 5

// ---------------- degree / normalization ----------------
__global__ void k_deg_init(float* __restrict__ deg) {
    int i = blockIdx.x * blockDim.x + threadIdx.x;
    if (i < N_NODES) deg[i] = 1.0f;   // self-loop contributes 1
}

__global__ void k_deg_accum(const int* __restrict__ edst, float* __restrict__ deg) {
    int e = blockIdx.x * blockDim.x + threadIdx.x;
    int stride = gridDim.x * blockDim.x;
    for (; e < N_EDGES; e += stride)
        atomicAdd(&deg[edst[e]], 1.0f);
}

__global__ void k_dinv(float* __restrict__ deg) {
    int i = blockIdx.x * blockDim.x + threadIdx.x;
    if (i < N_NODES) deg[i] = rsqrtf(deg[i]);   // deg >= 1 always
}

// ---------------- h = x @ W via V_WMMA_F32_16X16X4_F32 + TDM A-tile staging ----------------
// Block: 256 threads = 8 waves, owns an 80-row M strip. Wave w owns columns
// [16w, 16w+16): preloads its 128x16 B strip into 64 VGPRs once, then sweeps
// 5 A-tiles (staged to LDS by the Tensor Data Mover with +2-dword row pad).
__global__ __launch_bounds__(256) void k_gemm(const float* __restrict__ X,
                                              const float* __restrict__ W,
                                              float* __restrict__ H) {
    __shared__ float ldsA[ROWS_PER_BLK * LDS_STRIDE];

    const int row0 = blockIdx.x * ROWS_PER_BLK;

    if (threadIdx.x < 32) {  // wave 0 issues the tensor DMA (TDM ignores EXEC)
        const uint64_t gaddr = (uint64_t)(uintptr_t)(X + (size_t)row0 * CH);
        const uint32_t laddr = (uint32_t)(uintptr_t)ldsA;  // LDS aperture: low 32 bits = LDS offset

        v4u g0;
        g0[0] = 1u;                                   // count=1, user mode, no gather
        g0[1] = laddr;                                // lds_addr
        g0[2] = (uint32_t)gaddr;                      // global_addr[31:0]
        g0[3] = (uint32_t)((gaddr >> 32) & 0x1FFFFFFu) | (2u << 30);  // addr[56:32] | type=2

        v8i g1;
        // data_size=4B(2), pad_enable, pad_interval=6 (128 dwords), pad_amount=1 (2 dwords)
        g1[0] = (int)((2u << 16) | (1u << 20) | (6u << 22) | (1u << 25));
        g1[1] = (int)(128u << 16);                    // tensor_dim0[15:0]=128
        g1[2] = (int)((uint32_t)ROWS_PER_BLK << 16);  // tensor_dim0 hi=0 | tensor_dim1[15:0]=80
        g1[3] = (int)(128u << 16);                    // tensor_dim1 hi=0 | tile_dim0=128
        g1[4] = ROWS_PER_BLK;                         // tile_dim1=80, tile_dim2=0
        g1[5] = 128;                                  // tensor_dim0_stride[31:0]=128
        g1[6] = 0;
        g1[7] = 0;

        v4i gz4 = {0, 0, 0, 0};
        v8i gz8 = {0, 0, 0, 0, 0, 0, 0, 0};
        __builtin_amdgcn_tensor_load_to_lds(g0, g1, gz4, gz4, gz8, 0);
        __builtin_amdgcn_s_wait_tensorcnt(0);
    }
    __syncthreads();

    const int lane  = threadIdx.x & 31;
    const int wave  = threadIdx.x >> 5;
    const int col0  = wave * 16;
    const int lhalf = lane >> 4;     // 0: lanes 0-15, 1: lanes 16-31
    const int lcol  = lane & 15;     // M index for A, N index for B/C

    // Preload B strip: 32 fragments of 4x16 (per-lane v2f) = 64 VGPRs, reused for 5 tiles.
    v2f barr[32];
#pragma unroll
    for (int k = 0; k < 32; ++k) {
        const int k0 = 4 * k + 2 * lhalf;
        barr[k].x = W[(k0    ) * CH + col0 + lcol];
        barr[k].y = W[(k0 + 1) * CH + col0 + lcol];
    }

#pragma unroll
    for (int t = 0; t < M_TILES; ++t) {
        const int trow = t * 16;
        v8f acc = {0.f, 0.f, 0.f, 0.f, 0.f, 0.f, 0.f, 0.f};
#pragma unroll
        for (int k = 0; k < 32; ++k) {
            const int k0 = 4 * k + 2 * lhalf;
            v2f a;   // A 16x4: lanes 0-15 -> K=4k+0/1, lanes 16-31 -> K=4k+2/3, M=lcol
            a.x = ldsA[(trow + lcol) * LDS_STRIDE + k0];
            a.y = ldsA[(trow + lcol) * LDS_STRIDE + k0 + 1];
            acc = __builtin_amdgcn_wmma_f32_16x16x4_f32(
                false, a, false, barr[k], (short)0, acc, false, false);
        }
#pragma unroll
        for (int r = 0; r < 8; ++r)
            H[(size_t)(row0 + trow + r + 8 * lhalf) * CH + col0 + lcol] = acc[r];
    }
}

// ---------------- agg = h * dinv^2 + bias  (self-loop folded in) ----------------
__global__ void k_init_agg(const float* __restrict__ H, const float* __restrict__ dinv,
                           const float* __restrict__ bias, float* __restrict__ agg) {
    int idx = blockIdx.x * blockDim.x + threadIdx.x;
    int stride = gridDim.x * blockDim.x;
    const int total = N_NODES * CH;
    for (; idx < total; idx += stride) {
        int v = idx >> 7;
        int c = idx & 127;
        float d = dinv[v];
        agg[idx] = H[idx] * d * d + bias[c];
    }
}

// ---------------- edge scatter: agg[dst] += h[src] * dinv[src]*dinv[dst] ----------------
__global__ __launch_bounds__(256) void k_scatter(const float* __restrict__ H,
                                                 const float* __restrict__ dinv,
                                                 const int* __restrict__ esrc,
                                                 const int* __restrict__ edst,
                                                 float* __restrict__ agg) {
    const int lane = threadIdx.x & 31;
    int wid = blockIdx.x * (blockDim.x >> 5) + (threadIdx.x >> 5);
    const int nwaves = gridDim.x * (blockDim.x >> 5);
    for (int e = wid; e < N_EDGES; e += nwaves) {
        const int s = esrc[e];
        const int d = edst[e];
        const float norm = dinv[s] * dinv[d];
        float4 hv = ((const float4*)(H + (size_t)s * CH))[lane];
        float* ap = agg + (size_t)d * CH + lane * 4;
        atomicAdd(ap + 0, hv.x * norm);
        atomicAdd(ap + 1, hv.y * norm);
        atomicAdd(ap + 2, hv.z * norm);
        atomicAdd(ap + 3, hv.w * norm);
    }
}

// ---------------- x_out = relu(agg) + x_in ----------------
__global__ void k_update(const float* __restrict__ agg, const float* __restrict__ xin,
                         float* __restrict__ xout) {
    int idx = blockIdx.x * blockDim.x + threadIdx.x;
    int stride = gridDim.x * blockDim.x;
    const int total = N_NODES * CH;
    for (; idx < total; idx += stride)
        xout[idx] = fmaxf(agg[idx], 0.0f) + xin[idx];
}

extern "C" void kernel_launch(void* const* d_in, const int* in_sizes, int n_in,
                              void* d_out, int out_size, void* d_ws, size_t ws_size,
                              hipStream_t stream) {
    const float* x  = (const float*)d_in[0];            // [N, 128]
    const float* W  = (const float*)d_in[1];            // [3, 128, 128]
    const float* b  = (const float*)d_in[2];            // [3, 128]
    const int*   ei = (const int*)d_in[3];              // [2, E] (int per harness contract)
    const int*   esrc = ei;
    const int*   edst = ei + N_EDGES;
    float* out = (float*)d_out;

    // workspace layout (all 256B aligned)
    char* ws = (char*)d_ws;
    size_t off = 0;
    auto alloc = [&](size_t bytes) { char* p = ws + off; off = (off + bytes + 255) & ~(size_t)255; return p; };
    float* dinv = (float*)alloc(sizeof(float) * N_NODES);
    float* xbuf = (float*)alloc(sizeof(float) * N_NODES * CH);
    float* h    = (float*)alloc(sizeof(float) * N_NODES * CH);
    float* agg  = (float*)alloc(sizeof(float) * N_NODES * CH);
    (void)ws_size; (void)n_in; (void)in_sizes; (void)out_size;

    // normalization: deg -> dinv
    k_deg_init <<<(N_NODES + 255) / 256, 256, 0, stream>>>(dinv);
    k_deg_accum<<<1024, 256, 0, stream>>>(edst, dinv);
    k_dinv     <<<(N_NODES + 255) / 256, 256, 0, stream>>>(dinv);

    const float* xcur = x;
    for (int layer = 0; layer < 3; ++layer) {
        const float* Wl = W + (size_t)layer * CH * CH;
        const float* bl = b + (size_t)layer * CH;
        float* xnext = (layer == 2) ? out : xbuf;

        k_gemm    <<<N_NODES / ROWS_PER_BLK, 256, 0, stream>>>(xcur, Wl, h);
        k_init_agg<<<4096, 256, 0, stream>>>(h, dinv, bl, agg);
        k_scatter <<<2048, 256, 0, stream>>>(h, dinv, esrc, edst, agg);
        k_update  <<<4096, 256, 0, stream>>>(agg, xcur, xnext);

        xcur = xnext;
    }
}